// PRNN_867583394149
// MI455X (gfx1250) — compile-verified
//
#include <hip/hip_runtime.h>

// PRNN J2-plasticity scan, fused single kernel for gfx1250 (MI455X).
//
// Design rationale (compile-only; reasoned from MI455X specs):
//  - Total work ~0.7 GFLOP, real HBM traffic ~3 MB -> serial-scan /
//    VALU-latency bound, NOT bandwidth or tensor bound. So: fuse everything,
//    never spill the [B,S,192] strain tensor (100 MB) to HBM.
//  - Encoder [64s x 192l] per batch element is matrix-shaped -> compute once
//    per block with 48x V_WMMA_F32_16X16X4_F32 (K=3 zero-padded to 4),
//    staged to LDS. The 64-step serial loop then has zero encoder MACs and
//    zero barriers.
//  - Decoder 64->3 reduction per step via wave32 __shfl_xor butterflies,
//    wave partials combined once after the loop (single __syncthreads).

typedef __attribute__((ext_vector_type(2))) float v2f;
typedef __attribute__((ext_vector_type(8))) float v8f;

#define B_   2048
#define S_   64
#define M_   64
#define F_   3
#define O_   3
#define L_   192   // M_*F_
#define LSTR 196   // LDS row stride (floats), padded: 8*196 % 64 == 32 -> no
                   // write-bank overlap between lane halves of a WMMA tile

__global__ __launch_bounds__(64) void prnn_fused_kernel(
    const float* __restrict__ x,        // [B,S,F]
    const float* __restrict__ enc_w,    // [F,L]
    const float* __restrict__ dec_raw,  // [M,O]
    const float* __restrict__ material, // [4] = E, nu, sy0, H
    float* __restrict__ out)            // [B,S,O]
{
    __shared__ float lds_strain[S_ * LSTR];   // [s][l] encoder output
    __shared__ float lds_part[2][S_ * O_];    // per-wave decoder partials
    __shared__ float lds_sp[M_ * O_];         // softplus(dec_raw)

    const int t    = threadIdx.x;   // 0..63 == material point m
    const int b    = blockIdx.x;    // batch element
    const int wave = t >> 5;
    const int lane = t & 31;

    // ---------------- Phase 1a: encoder GEMM via WMMA, stage to LDS -------
    // strains[s,l] = sum_f x[b,s,f] * enc_w[f,l]
    // Tile rows = s (16), cols = l (16), K = f (3, padded to 4 with zeros).
    // A 16x4 f32 layout: lanes 0-15 -> M rows, VGPR0 = {K0 | K2}, VGPR1 = {K1 | K3}.
    // B 4x16 mirrors the same K-slot enumeration, so consistent placement of
    // (x_f, enc_w row f) in matching slots is correct for any hw K ordering.
    {
        const int hi = (lane < 16) ? 0 : 1;  // lane half
        const int lr = lane & 15;
        for (int sti = 0; sti < 2; ++sti) {  // each wave: 2 s-tiles of 16
            const int s0 = (wave * 2 + sti) * 16;
            const float* xr = x + ((size_t)b * S_ + (s0 + lr)) * F_;
            v2f a;
            a.x = xr[hi ? 2 : 0];            // K0 (lanes 0-15) / K2 (16-31)
            a.y = hi ? 0.0f : xr[1];         // K1 / K3(=0, pad)
            for (int lt = 0; lt < 12; ++lt) {
                const int l0 = lt * 16;
                v2f bm;
                bm.x = enc_w[(hi ? 2 : 0) * L_ + l0 + lr];
                bm.y = hi ? 0.0f : enc_w[1 * L_ + l0 + lr];
                v8f c = {};
                c = __builtin_amdgcn_wmma_f32_16x16x4_f32(
                        /*neg_a=*/false, a, /*neg_b=*/false, bm,
                        /*c_mod=*/(short)0, c, /*reuse_a=*/false, /*reuse_b=*/false);
                // C layout: lane half selects rows r / r+8, col = lane&15
                #pragma unroll
                for (int r = 0; r < 8; ++r) {
                    const int row = r + hi * 8;
                    lds_strain[(s0 + row) * LSTR + l0 + lr] = c[r];
                }
            }
        }
    }

    // ---------------- Phase 1b: softplus(dec_raw) to LDS -------------------
    #pragma unroll
    for (int o = 0; o < O_; ++o) {
        const float v = dec_raw[t * O_ + o];
        lds_sp[t * O_ + o] = fmaxf(v, 0.0f) + log1pf(expf(-fabsf(v)));
    }

    __syncthreads();   // strains + softplus staged

    // Column-normalized decoder weights for this material point
    float w0, w1, w2;
    {
        float cs0 = 0.f, cs1 = 0.f, cs2 = 0.f;
        for (int m = 0; m < M_; ++m) {
            cs0 += lds_sp[m * O_ + 0];
            cs1 += lds_sp[m * O_ + 1];
            cs2 += lds_sp[m * O_ + 2];
        }
        w0 = lds_sp[t * O_ + 0] / cs0;
        w1 = lds_sp[t * O_ + 1] / cs1;
        w2 = lds_sp[t * O_ + 2] / cs2;
    }

    // ---------------- Phase 2: sequential J2 scan (barrier-free) -----------
    const float E   = material[0];
    const float nu  = material[1];
    const float sy0 = material[2];
    const float H   = material[3];
    const float mu    = E / (2.0f * (1.0f + nu));
    const float lam   = E * nu / ((1.0f + nu) * (1.0f - 2.0f * nu));
    const float twomu = 2.0f * mu;
    const float inv_d = 1.0f / (3.0f * mu + H);

    float ep0 = 0.f, ep1 = 0.f, ep2 = 0.f, ep3 = 0.f, ebar = 0.f;

    for (int s = 0; s < S_; ++s) {
        const float st0 = lds_strain[s * LSTR + t * 3 + 0];
        const float st1 = lds_strain[s * LSTR + t * 3 + 1];
        const float st2 = lds_strain[s * LSTR + t * 3 + 2];

        const float exx = st0 - ep0;
        const float eyy = st1 - ep1;
        const float ezz = -ep2;
        const float exy = 0.5f * st2 - ep3;
        const float tr  = exx + eyy + ezz;
        const float sxx = lam * tr + twomu * exx;
        const float syy = lam * tr + twomu * eyy;
        const float szz = lam * tr + twomu * ezz;
        const float sxy = twomu * exy;
        const float p   = (sxx + syy + szz) * (1.0f / 3.0f);
        const float dxx = sxx - p, dyy = syy - p, dzz = szz - p;
        const float q   = sqrtf(1.5f * (dxx * dxx + dyy * dyy + dzz * dzz
                                        + 2.0f * sxy * sxy));
        const float fy   = q - (sy0 + H * ebar);
        const float dgam = (fy > 0.0f) ? fy * inv_d : 0.0f;
        const float sq   = (q > 1e-12f) ? q : 1.0f;
        const float rq   = 1.5f / sq;          // single divide per step

        ep0 += dgam * (dxx * rq);
        ep1 += dgam * (dyy * rq);
        ep2 += dgam * (dzz * rq);
        ep3 += dgam * (sxy * rq);
        ebar += dgam;

        const float scale = 1.0f - twomu * dgam * rq;  // == 1 - 3mu*dgam/q
        float r0 = (scale * dxx + p) * w0;
        float r1 = (scale * dyy + p) * w1;
        float r2 = (scale * sxy)     * w2;

        // wave32 butterfly reduction over 32 material points
        #pragma unroll
        for (int off = 16; off > 0; off >>= 1) {
            r0 += __shfl_xor(r0, off, 32);
            r1 += __shfl_xor(r1, off, 32);
            r2 += __shfl_xor(r2, off, 32);
        }
        if (lane == 0) {
            lds_part[wave][s * 3 + 0] = r0;
            lds_part[wave][s * 3 + 1] = r1;
            lds_part[wave][s * 3 + 2] = r2;
        }
    }

    __syncthreads();

    // ---------------- Phase 3: combine wave partials, write out ------------
    float* ob = out + (size_t)b * (S_ * O_);
    for (int idx = t; idx < S_ * O_; idx += 64)
        ob[idx] = lds_part[0][idx] + lds_part[1][idx];
}

extern "C" void kernel_launch(void* const* d_in, const int* in_sizes, int n_in,
                              void* d_out, int out_size, void* d_ws, size_t ws_size,
                              hipStream_t stream) {
    (void)in_sizes; (void)n_in; (void)d_ws; (void)ws_size; (void)out_size;
    const float* x        = (const float*)d_in[0];
    const float* enc_w    = (const float*)d_in[1];
    const float* dec_raw  = (const float*)d_in[2];
    const float* material = (const float*)d_in[3];
    float* out = (float*)d_out;
    prnn_fused_kernel<<<B_, 64, 0, stream>>>(x, enc_w, dec_raw, material, out);
}